// LocalMemoryModule_54434415509784
// MI455X (gfx1250) — compile-verified
//
#include <hip/hip_runtime.h>
#include <hip/hip_bf16.h>
#include <math.h>

// ---------------------------------------------------------------------------
// Problem constants (from reference setup_inputs)
// ---------------------------------------------------------------------------
constexpr int Bn  = 16;    // batch
constexpr int Tn  = 12;    // time steps
constexpr int Nn  = 1024;  // nodes
constexpr int Dn  = 128;   // feature dim
constexpr int Fn  = 8;     // raw features
constexpr int TAU = 4;
constexpr int KNB = 16;
constexpr int ROWS = Bn * Nn;       // 16384 rows for all dense GEMMs
constexpr int MTILE = 64;           // KNB * TAU attention entries per node

typedef __attribute__((ext_vector_type(2))) float    v2f;
typedef __attribute__((ext_vector_type(8))) float    v8f;
typedef __attribute__((ext_vector_type(4))) unsigned u32x4;
typedef __attribute__((ext_vector_type(8))) unsigned u32x8;

// ---------------------------------------------------------------------------
// Kernel 0: fuse wqk = wk^T @ wq  and  bqk = wk^T @ bq
// (scores[m] = (wk^T q) . nb[m]; the q.bk term is softmax-invariant -> dropped)
// ---------------------------------------------------------------------------
__global__ __launch_bounds__(Dn) void k_wqk(const float* __restrict__ wq,
                                            const float* __restrict__ bq,
                                            const float* __restrict__ wk,
                                            float* __restrict__ wqk,
                                            float* __restrict__ bqk) {
  const int j = blockIdx.x;
  const int d = threadIdx.x;
  float acc = 0.f;
  for (int p = 0; p < Dn; ++p)
    acc += wk[p * Dn + j] * wq[p * Dn + d];
  wqk[j * Dn + d] = acc;
  if (d == 0) {
    float s = 0.f;
    for (int p = 0; p < Dn; ++p) s += wk[p * Dn + j] * bq[p];
    bqk[j] = s;
  }
}

// ---------------------------------------------------------------------------
// Kernel 1: KNN top-16 by squared distance on wind = x_orig[T-1, b, n*8+4:6].
// Strict '<' insertion reproduces jax.lax.top_k ordering (ascending distance,
// lower index first on ties).
// ---------------------------------------------------------------------------
__global__ __launch_bounds__(256) void k_topk(const float* __restrict__ x_orig,
                                              int* __restrict__ topk) {
  __shared__ float sx[Nn];
  __shared__ float sy[Nn];
  const int node = blockIdx.x * 256 + threadIdx.x;
  const int b = node >> 10;                 // uniform within the block
  const float* xb = x_orig + (size_t)((Tn - 1) * Bn + b) * (Nn * Fn);
  for (int j = threadIdx.x; j < Nn; j += 256) {
    sx[j] = xb[j * Fn + 4];
    sy[j] = xb[j * Fn + 5];
  }
  __syncthreads();

  const int n = node & (Nn - 1);
  const float px = sx[n], py = sy[n];

  float bd[KNB];
  int   bi[KNB];
#pragma unroll
  for (int p = 0; p < KNB; ++p) { bd[p] = INFINITY; bi[p] = 0; }

  for (int j = 0; j < Nn; ++j) {
    const float dx = px - sx[j];
    const float dy = py - sy[j];
    const float d2 = dx * dx + dy * dy;
    if (d2 < bd[KNB - 1]) {
      bd[KNB - 1] = d2; bi[KNB - 1] = j;
#pragma unroll
      for (int p = KNB - 1; p >= 1; --p) {
        if (bd[p] < bd[p - 1]) {
          float td = bd[p]; bd[p] = bd[p - 1]; bd[p - 1] = td;
          int   ti = bi[p]; bi[p] = bi[p - 1]; bi[p - 1] = ti;
        }
      }
    }
  }
#pragma unroll
  for (int p = 0; p < KNB; ++p) topk[node * KNB + p] = bi[p];
}

// ---------------------------------------------------------------------------
// Kernel 2: dense GEMM  Y[r, j] = sum_d X[r, d] * W[j, d] + bias[j]
// V_WMMA_F32_16X16X4_F32, one wave -> 64x16 output tile: four 16x16
// accumulators share one B fragment per k-step (4x less B traffic, 4 WMMAs
// per B load).  fp32 end-to-end.
// ---------------------------------------------------------------------------
__global__ __launch_bounds__(128) void k_gemm_wmma(const float* __restrict__ X,
                                                   long long xbs,
                                                   const float* __restrict__ W,
                                                   const float* __restrict__ bias,
                                                   float* __restrict__ Y,
                                                   int act) {
  const int wave = (blockIdx.x << 2) + (threadIdx.x >> 5);  // 2048 waves total
  const int lane = threadIdx.x & 31;
  const int half = lane >> 4;     // 0 or 1
  const int lm   = lane & 15;
  const int i0 = (wave >> 3) << 6;         // 64-row tile (256 row tiles)
  const int j0 = (wave & 7) << 4;          // 16-col tile

  const int r0 = i0 + lm;                  // whole 64-row tile shares (r0>>10)
  const float* xbase = X + (size_t)(r0 >> 10) * (size_t)xbs
                         + (size_t)(r0 & 1023) * Dn + 2 * half;
  const float* wrow  = W + (size_t)(j0 + lm) * Dn + 2 * half;
  __builtin_prefetch(wrow, 0, 3);          // global_prefetch of the B tile

  const float bj = bias[j0 + lm];
  v8f c0, c1, c2, c3;
#pragma unroll
  for (int r = 0; r < 8; ++r) { c0[r] = bj; c1[r] = bj; c2[r] = bj; c3[r] = bj; }

#pragma unroll 4
  for (int kk = 0; kk < 32; ++kk) {
    const v2f bb = *(const v2f*)(wrow + kk * 4);
    const v2f a0 = *(const v2f*)(xbase +  0 * Dn + kk * 4);
    const v2f a1 = *(const v2f*)(xbase + 16 * Dn + kk * 4);
    const v2f a2 = *(const v2f*)(xbase + 32 * Dn + kk * 4);
    const v2f a3 = *(const v2f*)(xbase + 48 * Dn + kk * 4);
    c0 = __builtin_amdgcn_wmma_f32_16x16x4_f32(false, a0, false, bb, (short)0, c0, false, false);
    c1 = __builtin_amdgcn_wmma_f32_16x16x4_f32(false, a1, false, bb, (short)0, c1, false, false);
    c2 = __builtin_amdgcn_wmma_f32_16x16x4_f32(false, a2, false, bb, (short)0, c2, false, false);
    c3 = __builtin_amdgcn_wmma_f32_16x16x4_f32(false, a3, false, bb, (short)0, c3, false, false);
  }

#pragma unroll
  for (int s = 0; s < 4; ++s) {
    const v8f c = (s == 0) ? c0 : (s == 1) ? c1 : (s == 2) ? c2 : c3;
#pragma unroll
    for (int r = 0; r < 8; ++r) {
      float v = c[r];
      if (act) v = 0.5f * v * (1.0f + erff(v * 0.70710678118654752440f)); // exact GELU
      Y[(size_t)(i0 + 16 * s + r + 8 * half) * Dn + (j0 + lm)] = v;
    }
  }
}

// ---------------------------------------------------------------------------
// Kernel 3: fused gather + attention per node, gather done by the
// Tensor Data Mover in GATHER mode (one tensor_load_to_lds per history step:
// 16 rows of 128 fp32 selected by 16-bit row indices from h_e[b, t', :, :]).
//   scores[m=kk*4+tp] = (qk . nb) / sqrt(D);  attn = softmax
//   u = sum_m attn[m] * nb[m,:]   (ctx = wv @ u + bv done by the GEMM after)
// TDM destination is s_nb[tp][kk][d] so the m -> (kk, tp) mapping matches the
// reference's reshape order.
// ---------------------------------------------------------------------------
__global__ __launch_bounds__(64) void k_attn(const float* __restrict__ h_e,
                                             const float* __restrict__ qk,
                                             const int* __restrict__ topk,
                                             float* __restrict__ u) {
  __shared__ float s_qk[Dn];
  __shared__ float s_nb[TAU][KNB][Dn];     // TDM gather lands here (row-major)
  __shared__ float s_sc[MTILE];
  __shared__ float s_at[MTILE];

  const int node = blockIdx.x;
  const int b = node >> 10;
  const int m = threadIdx.x;               // 0..63

  s_qk[m]      = qk[(size_t)node * Dn + m];
  s_qk[m + 64] = qk[(size_t)node * Dn + m + 64];

  if (m < 32) {  // wave 0 issues the 4 TDM gathers (EXEC ignored by TDM)
    // --- pack 16 neighbor indices as 16-bit row indices (D# groups 2+3) ---
    const int* tk = topk + node * KNB;
    unsigned pk[8];
#pragma unroll
    for (int i = 0; i < 8; ++i) {
      const unsigned lo = (unsigned)tk[2 * i];
      const unsigned hi = (unsigned)tk[2 * i + 1];
      pk[i] = (unsigned)__builtin_amdgcn_readfirstlane((int)(lo | (hi << 16)));
    }
    const u32x4 g2 = { pk[0], pk[1], pk[2], pk[3] };
    const u32x4 g3 = { pk[4], pk[5], pk[6], pk[7] };

    // --- D# group1: data_size=4B, tensor_dim0=128, tensor_dim1=1024,
    //     tile_dim0=128, tile_dim1=16 indices, dim0_stride=128 ---
    const u32x8 g1 = { 0x00020000u,          // data_size=2 (4B), no pad/iterate
                       (unsigned)(Dn << 16), // tensor_dim0 low16 in [31:16]
                       (unsigned)(Nn << 16), // tensor_dim1 low16 in [31:16]
                       (unsigned)(Dn << 16), // tile_dim0 in [31:16]
                       (unsigned)KNB,        // tile_dim1 = #indices
                       (unsigned)Dn,         // tensor_dim0_stride low32
                       0u, 0u };

#pragma unroll
    for (int tp = 0; tp < TAU; ++tp) {
      const unsigned long long ga =
          (unsigned long long)(const void*)(h_e + (size_t)(b * Tn + (Tn - TAU) + tp) * Nn * Dn);
      const unsigned ldsa = (unsigned)(size_t)(&s_nb[tp][0][0]); // low 32 bits = LDS offset
      const u32x4 g0 = { 0x80000001u,                        // gather_mode=1, count=1
                         ldsa,                               // lds_addr
                         (unsigned)ga,                       // global_addr[31:0]
                         (unsigned)(ga >> 32) | 0x80000000u  // addr[56:32] | type=2
                       };
      asm volatile("tensor_load_to_lds %0, %1, %2, %3"
                   :: "s"(g0), "s"(g1), "s"(g2), "s"(g3)
                   : "memory");
    }
    __builtin_amdgcn_s_wait_tensorcnt(0);
  }
  __syncthreads();

  // --- scores: dot(qk, nb[m]) with lane-skewed LDS reads (bank-conflict-free)
  const int kk = m >> 2, tp = m & 3;
  const float* row = &s_nb[tp][kk][0];
  float dot = 0.f;
#pragma unroll 4
  for (int i = 0; i < Dn; ++i) {
    const int d = (i + 2 * m) & (Dn - 1);
    dot += s_qk[d] * row[d];
  }
  s_sc[m] = dot * 0.08838834764831845f;    // 1/sqrt(128)
  __syncthreads();

  float mx = -INFINITY;
  for (int i = 0; i < MTILE; ++i) mx = fmaxf(mx, s_sc[i]);
  s_at[m] = expf(s_sc[m] - mx);
  __syncthreads();

  float sum = 0.f;
  for (int i = 0; i < MTILE; ++i) sum += s_at[i];
  const float inv = 1.0f / sum;

  float acc0 = 0.f, acc1 = 0.f;
  for (int i = 0; i < MTILE; ++i) {
    const float a = s_at[i];
    acc0 += a * s_nb[i & 3][i >> 2][m];
    acc1 += a * s_nb[i & 3][i >> 2][m + 64];
  }
  u[(size_t)node * Dn + m]      = acc0 * inv;
  u[(size_t)node * Dn + m + 64] = acc1 * inv;
}

// ---------------------------------------------------------------------------
// Launch
// ---------------------------------------------------------------------------
extern "C" void kernel_launch(void* const* d_in, const int* in_sizes, int n_in,
                              void* d_out, int out_size, void* d_ws, size_t ws_size,
                              hipStream_t stream) {
  const float* h_e    = (const float*)d_in[0];
  const float* x_orig = (const float*)d_in[1];
  const float* wq     = (const float*)d_in[2];
  const float* bq     = (const float*)d_in[3];
  const float* wk     = (const float*)d_in[4];
  // d_in[5] = bk: q.bk is constant over m -> softmax-invariant, unused
  const float* wv     = (const float*)d_in[6];
  const float* bv     = (const float*)d_in[7];
  const float* w1     = (const float*)d_in[8];
  const float* b1     = (const float*)d_in[9];
  const float* w2     = (const float*)d_in[10];
  const float* b2     = (const float*)d_in[11];
  float* out = (float*)d_out;

  // Workspace (~17.9 MB): wqk | bqk | buf0 (qk->ctx) | buf1 (u->h1) | topk
  float* ws   = (float*)d_ws;
  float* wqk  = ws;
  float* bqk  = wqk + Dn * Dn;
  float* buf0 = bqk + Dn;
  float* buf1 = buf0 + (size_t)ROWS * Dn;
  int*   topk = (int*)(buf1 + (size_t)ROWS * Dn);

  const int gemm_blocks = (ROWS / 64) * (Dn / 16) / 4;   // 2048 waves / 4 per block

  k_wqk<<<Dn, Dn, 0, stream>>>(wq, bq, wk, wqk, bqk);
  k_topk<<<ROWS / 256, 256, 0, stream>>>(x_orig, topk);

  // qk = h_e[:, T-1] @ wqk^T + bqk  (reads t=T-1 slice of h_e in place)
  k_gemm_wmma<<<gemm_blocks, 128, 0, stream>>>(
      h_e + (size_t)(Tn - 1) * Nn * Dn, (long long)Tn * Nn * Dn,
      wqk, bqk, buf0, /*act=*/0);

  // TDM gather + attention -> u
  k_attn<<<ROWS, 64, 0, stream>>>(h_e, buf0, topk, buf1);

  // ctx = u @ wv^T + bv
  k_gemm_wmma<<<gemm_blocks, 128, 0, stream>>>(
      buf1, (long long)Nn * Dn, wv, bv, buf0, /*act=*/0);

  // h = gelu(ctx @ w1^T + b1)
  k_gemm_wmma<<<gemm_blocks, 128, 0, stream>>>(
      buf0, (long long)Nn * Dn, w1, b1, buf1, /*act=*/1);

  // out = h @ w2^T + b2
  k_gemm_wmma<<<gemm_blocks, 128, 0, stream>>>(
      buf1, (long long)Nn * Dn, w2, b2, out, /*act=*/0);
}